// GaANConv_38397007626856
// MI455X (gfx1250) — compile-verified
//
#include <hip/hip_runtime.h>

typedef float v2f __attribute__((ext_vector_type(2)));
typedef float v8f __attribute__((ext_vector_type(8)));

#define LANE() ((int)(threadIdx.x & 31))
#define WAVE() ((int)(threadIdx.x >> 5))

// ---------------------------------------------------------------------------
// f32 WMMA 16x16x4 helpers (CDNA5 wave32 layouts per CDNA5 ISA):
//  A 16x4 : lane l holds row m=l&15; lanes 0-15 -> K={0,1}, lanes 16-31 -> K={2,3}
//  B 4x16 : lane l holds col n=l&15; lanes 0-15 -> K={0,1}, lanes 16-31 -> K={2,3}
//  C/D    : reg r, lane l -> (row = r + 8*(l>>4), col = l&15)
// ---------------------------------------------------------------------------
__device__ __forceinline__ v8f wmma_f32_16x16x4(v2f a, v2f b, v8f c) {
  return __builtin_amdgcn_wmma_f32_16x16x4_f32(false, a, false, b, (short)0, c,
                                               false, false);
}

// C(16x16) = A(16xK from LDS, row stride sa) * B(Kx64 row-major global, cols j0..j0+15)
__device__ __forceinline__ v8f gemm_tile_f32(const float* __restrict__ sA, int sa,
                                             const float* __restrict__ B, int ldb,
                                             int j0, int K) {
  const int lane = LANE();
  const int mn = lane & 15;      // A row == B column within tile
  const int ksel = lane >> 4;    // which K pair inside each K=4 step
  v8f acc = {};
  for (int kk = 0; kk < K; kk += 4) {
    const int ka = kk + 2 * ksel;
    v2f a, b;
    a.x = sA[mn * sa + ka];
    a.y = sA[mn * sa + ka + 1];
    b.x = B[ka * ldb + j0 + mn];
    b.y = B[(ka + 1) * ldb + j0 + mn];
    acc = wmma_f32_16x16x4(a, b, acc);
  }
  return acc;
}

__device__ __forceinline__ void store_tile_global(float* __restrict__ out, int ldo,
                                                  int row0, int j0, v8f acc, int nrows) {
  const int lane = LANE();
  const int n = lane & 15;
#pragma unroll
  for (int r = 0; r < 8; ++r) {
    const int row = row0 + r + 8 * (lane >> 4);
    if (row < nrows) out[(long long)row * ldo + j0 + n] = acc[r];
  }
}

__device__ __forceinline__ void store_tile_lds(float* sC, int sc, int j0, v8f acc) {
  const int lane = LANE();
  const int n = lane & 15;
#pragma unroll
  for (int r = 0; r < 8; ++r) {
    const int tr = r + 8 * (lane >> 4);
    sC[tr * sc + j0 + n] = acc[r];
  }
}

// ---------------------------------------------------------------------------
// Ordered float <-> uint map for atomic segment-max (monotone bijection).
// amax buffer zero-init == mapped(-NaN) => below every real value.
// ---------------------------------------------------------------------------
__device__ __forceinline__ unsigned f32_to_ordered_u32(float f) {
  unsigned b = __float_as_uint(f);
  return (b & 0x80000000u) ? ~b : (b | 0x80000000u);
}
__device__ __forceinline__ float ordered_u32_to_f32(unsigned u) {
  unsigned b = (u & 0x80000000u) ? (u & 0x7FFFFFFFu) : ~u;
  return __uint_as_float(b);
}

// ---------------------------------------------------------------------------
// W1 = Wpi @ Wa[0:64,:], W2 = Wpj @ Wa[64:128,:], bAll = bpi@WaTop + bpj@WaMid + ba
// ---------------------------------------------------------------------------
__global__ __launch_bounds__(256) void fuse_weights_kernel(
    const float* __restrict__ Wpi, const float* __restrict__ bpi,
    const float* __restrict__ Wpj, const float* __restrict__ bpj,
    const float* __restrict__ Wa, const float* __restrict__ ba,
    float* __restrict__ W1, float* __restrict__ W2, float* __restrict__ bAll) {
  const int h = blockIdx.x;
  const float* wpi = Wpi + (size_t)h * 64 * 64;
  const float* wpj = Wpj + (size_t)h * 64 * 64;
  const float* waT = Wa + (size_t)h * 144 * 64;  // rows 0..63
  const float* waM = waT + 64 * 64;              // rows 64..127
  float* w1 = W1 + (size_t)h * 64 * 64;
  float* w2 = W2 + (size_t)h * 64 * 64;
  for (int f = threadIdx.x; f < 64 * 64; f += 256) {
    const int i = f >> 6, j = f & 63;
    float s1 = 0.f, s2 = 0.f;
    for (int k = 0; k < 64; ++k) {
      s1 += wpi[i * 64 + k] * waT[k * 64 + j];
      s2 += wpj[i * 64 + k] * waM[k * 64 + j];
    }
    w1[f] = s1;
    w2[f] = s2;
  }
  if (threadIdx.x < 64) {
    const int j = threadIdx.x;
    float s = ba[h * 64 + j];
    for (int k = 0; k < 64; ++k) {
      s += bpi[h * 64 + k] * waT[k * 64 + j];
      s += bpj[h * 64 + k] * waM[k * 64 + j];
    }
    bAll[h * 64 + j] = s;
  }
}

// ---------------------------------------------------------------------------
// Per head: x_fc = x@Wn + bn ; u = x_fc@W1 + bAll ; v = x_fc@W2
// Block = 4 waves, each wave owns a 16-node tile. WMMA f32.
// ---------------------------------------------------------------------------
__global__ __launch_bounds__(128) void node_transform_kernel(
    const float* __restrict__ x, const float* __restrict__ Wn,
    const float* __restrict__ bn, const float* __restrict__ W1,
    const float* __restrict__ W2, const float* __restrict__ bAll,
    float* __restrict__ xfc, float* __restrict__ u, float* __restrict__ v, int N) {
  __shared__ float sX[4][16 * 132];  // padded: stride%64==4 -> conflict-free A fetch
  __shared__ float sF[4][16 * 66];   // x_fc tile, padded stride
  const int w = WAVE();
  const int lane = LANE();
  const int base = (blockIdx.x * 4 + w) * 16;
  const bool active = base < N;

  if (active) {
    for (int idx = lane; idx < 16 * 128; idx += 32) {
      const int r = idx >> 7, c = idx & 127;
      const int row = base + r;
      sX[w][r * 132 + c] = (row < N) ? x[(long long)row * 128 + c] : 0.0f;
    }
  }
  __syncthreads();

  if (active) {
    const int n = lane & 15;
#pragma unroll
    for (int ct = 0; ct < 4; ++ct) {
      const int j0 = ct * 16;
      v8f acc = gemm_tile_f32(sX[w], 132, Wn, 64, j0, 128);
      const float bc = bn[j0 + n];
#pragma unroll
      for (int r = 0; r < 8; ++r) acc[r] += bc;
      store_tile_lds(sF[w], 66, j0, acc);
      store_tile_global(xfc, 64, base, j0, acc, N);
    }
  }
  __syncthreads();

  if (active) {
    const int n = lane & 15;
#pragma unroll
    for (int ct = 0; ct < 4; ++ct) {
      const int j0 = ct * 16;
      v8f au = gemm_tile_f32(sF[w], 66, W1, 64, j0, 64);
      const float bc = bAll[j0 + n];
#pragma unroll
      for (int r = 0; r < 8; ++r) au[r] += bc;
      store_tile_global(u, 64, base, j0, au, N);
      v8f av = gemm_tile_f32(sF[w], 66, W2, 64, j0, 64);
      store_tile_global(v, 64, base, j0, av, N);
    }
  }
}

// ---------------------------------------------------------------------------
// Pass A: alpha[e,c] = leaky(u[dst,c] + v[src,c] + ea[e,:]@W3[:,c]); atomic seg-max
// One thread per (edge, channel). alpha is a 205MB/head stream -> non-temporal
// so the ~90MB of gathered node tables stay resident in the 192MB L2.
// ---------------------------------------------------------------------------
__global__ __launch_bounds__(256) void edge_alpha_kernel(
    const float* __restrict__ ea, const int* __restrict__ ei,
    const float* __restrict__ W3, const float* __restrict__ u,
    const float* __restrict__ v, float* __restrict__ alpha,
    unsigned* __restrict__ amax, int E) {
  __shared__ float sW3[16 * 64];
  for (int i = threadIdx.x; i < 16 * 64; i += 256) sW3[i] = W3[i];
  __syncthreads();
  const long long gid = (long long)blockIdx.x * 256 + threadIdx.x;
  if (gid >= (long long)E * 64) return;
  const int e = (int)(gid >> 6);
  const int c = (int)(gid & 63);
  const int src = ei[e];
  const int dst = ei[E + e];
  float a = u[(long long)dst * 64 + c] + v[(long long)src * 64 + c];
  const float4* __restrict__ ea4 = (const float4*)(ea + (long long)e * 16);
#pragma unroll
  for (int q = 0; q < 4; ++q) {
    const float4 t = ea4[q];
    a += t.x * sW3[(q * 4 + 0) * 64 + c];
    a += t.y * sW3[(q * 4 + 1) * 64 + c];
    a += t.z * sW3[(q * 4 + 2) * 64 + c];
    a += t.w * sW3[(q * 4 + 3) * 64 + c];
  }
  a = (a >= 0.f) ? a : 0.2f * a;  // leaky relu
  __builtin_nontemporal_store(a, &alpha[gid]);
  atomicMax(&amax[(long long)dst * 64 + c], f32_to_ordered_u32(a));
}

// Pass B: alpha <- exp(alpha - amax[dst]); atomic seg-sum into denom
__global__ __launch_bounds__(256) void edge_exp_kernel(
    const int* __restrict__ ei, const unsigned* __restrict__ amax,
    float* __restrict__ alpha, float* __restrict__ denom, int E) {
  const long long gid = (long long)blockIdx.x * 256 + threadIdx.x;
  if (gid >= (long long)E * 64) return;
  const int e = (int)(gid >> 6);
  const int c = (int)(gid & 63);
  const int dst = ei[E + e];
  float m = ordered_u32_to_f32(amax[(long long)dst * 64 + c]);
  if ((__float_as_uint(m) & 0x7F800000u) == 0x7F800000u) m = 0.0f;  // !isfinite -> 0
  const float a = __builtin_nontemporal_load(&alpha[gid]);
  const float ex = __expf(a - m);
  __builtin_nontemporal_store(ex, &alpha[gid]);
  atomicAdd(&denom[(long long)dst * 64 + c], ex);
}

// Pass C: hout[dst] += (alpha/denom[dst]) * x_fc[src]
__global__ __launch_bounds__(256) void edge_aggregate_kernel(
    const int* __restrict__ ei, const float* __restrict__ alpha,
    const float* __restrict__ denom, const float* __restrict__ xfc,
    float* __restrict__ hout, int E) {
  const long long gid = (long long)blockIdx.x * 256 + threadIdx.x;
  if (gid >= (long long)E * 64) return;
  const int e = (int)(gid >> 6);
  const int c = (int)(gid & 63);
  const int src = ei[e];
  const int dst = ei[E + e];
  const float a = __builtin_nontemporal_load(&alpha[gid]);
  const float wgt = a / (denom[(long long)dst * 64 + c] + 1e-16f);
  atomicAdd(&hout[(long long)dst * 64 + c], wgt * xfc[(long long)src * 64 + c]);
}

// ---------------------------------------------------------------------------
// out = [x | hout0..3] @ Wfc + bfc    (K = 384), WMMA f32, 2 waves/block
// ---------------------------------------------------------------------------
__global__ __launch_bounds__(64) void final_proj_kernel(
    const float* __restrict__ x, const float* __restrict__ hout,
    const float* __restrict__ Wfc, const float* __restrict__ bfc,
    float* __restrict__ out, int N) {
  __shared__ float sA[2][16 * 388];  // padded concat rows
  const int w = WAVE();
  const int lane = LANE();
  const int base = (blockIdx.x * 2 + w) * 16;
  const bool active = base < N;
  if (active) {
    for (int idx = lane; idx < 16 * 384; idx += 32) {
      const int r = idx / 384, cc = idx % 384;
      const int row = base + r;
      float val = 0.f;
      if (row < N) {
        if (cc < 128) {
          val = x[(long long)row * 128 + cc];
        } else {
          const int hh = (cc - 128) >> 6;
          const int c = (cc - 128) & 63;
          val = hout[((long long)hh * N + row) * 64 + c];
        }
      }
      sA[w][r * 388 + cc] = val;
    }
  }
  __syncthreads();
  if (active) {
    const int n = lane & 15;
#pragma unroll
    for (int ct = 0; ct < 4; ++ct) {
      const int j0 = ct * 16;
      v8f acc = gemm_tile_f32(sA[w], 388, Wfc, 64, j0, 384);
      const float bc = bfc[j0 + n];
#pragma unroll
      for (int r = 0; r < 8; ++r) acc[r] += bc;
      store_tile_global(out, 64, base, j0, acc, N);
    }
  }
}

__global__ void zero_kernel(float* __restrict__ p, long long n) {
  const long long i = (long long)blockIdx.x * blockDim.x + threadIdx.x;
  if (i < n) p[i] = 0.0f;
}

// ---------------------------------------------------------------------------
extern "C" void kernel_launch(void* const* d_in, const int* in_sizes, int n_in,
                              void* d_out, int out_size, void* d_ws, size_t ws_size,
                              hipStream_t stream) {
  (void)n_in; (void)out_size; (void)ws_size;
  const float* x   = (const float*)d_in[0];
  const int*   ei  = (const int*)d_in[1];
  const float* ea  = (const float*)d_in[2];
  const float* Wn  = (const float*)d_in[3];
  const float* bn  = (const float*)d_in[4];
  const float* Wpi = (const float*)d_in[5];
  const float* bpi = (const float*)d_in[6];
  const float* Wpj = (const float*)d_in[7];
  const float* bpj = (const float*)d_in[8];
  const float* Wa  = (const float*)d_in[9];
  const float* ba  = (const float*)d_in[10];
  const float* Wfc = (const float*)d_in[11];
  const float* bfc = (const float*)d_in[12];

  const int N = in_sizes[0] / 128;
  const int E = in_sizes[1] / 2;
  const int H = 4;

  float* ws = (float*)d_ws;
  size_t off = 0;
  float* W1   = ws + off; off += (size_t)H * 64 * 64;
  float* W2   = ws + off; off += (size_t)H * 64 * 64;
  float* bAll = ws + off; off += (size_t)H * 64;
  off = (off + 255) & ~(size_t)255;
  float*    xfc   = ws + off; off += (size_t)N * 64;
  float*    u     = ws + off; off += (size_t)N * 64;
  float*    v     = ws + off; off += (size_t)N * 64;
  unsigned* amax  = (unsigned*)(ws + off); off += (size_t)N * 64;  // adjacent to denom
  float*    denom = ws + off; off += (size_t)N * 64;
  float*    hout  = ws + off; off += (size_t)H * N * 64;
  float*    alpha = ws + off; off += (size_t)E * 64;

  fuse_weights_kernel<<<H, 256, 0, stream>>>(Wpi, bpi, Wpj, bpj, Wa, ba, W1, W2, bAll);

  {
    const long long n = (long long)H * N * 64;
    zero_kernel<<<(unsigned)((n + 255) / 256), 256, 0, stream>>>(hout, n);
  }

  const long long edgeThreads = (long long)E * 64;
  const unsigned edgeBlocks = (unsigned)((edgeThreads + 255) / 256);
  const unsigned nodeBlocks = (unsigned)((N + 63) / 64);

  for (int h = 0; h < H; ++h) {
    node_transform_kernel<<<nodeBlocks, 128, 0, stream>>>(
        x, Wn + (size_t)h * 128 * 64, bn + (size_t)h * 64,
        W1 + (size_t)h * 64 * 64, W2 + (size_t)h * 64 * 64,
        bAll + (size_t)h * 64, xfc, u, v, N);
    {
      const long long n = (long long)2 * N * 64;  // zero amax+denom (contiguous)
      zero_kernel<<<(unsigned)((n + 255) / 256), 256, 0, stream>>>((float*)amax, n);
    }
    edge_alpha_kernel<<<edgeBlocks, 256, 0, stream>>>(
        ea, ei, Wa + (size_t)h * 144 * 64 + 128 * 64, u, v, alpha, amax, E);
    edge_exp_kernel<<<edgeBlocks, 256, 0, stream>>>(ei, amax, alpha, denom, E);
    edge_aggregate_kernel<<<edgeBlocks, 256, 0, stream>>>(
        ei, alpha, denom, xfc, hout + (size_t)h * N * 64, E);
  }

  final_proj_kernel<<<(unsigned)((N + 31) / 32), 64, 0, stream>>>(
      x, hout, Wfc, bfc, (float*)d_out, N);
}